// AttentionBlock_78529182040813
// MI455X (gfx1250) — compile-verified
//
#include <hip/hip_runtime.h>
#include <math.h>

// ---- problem constants -----------------------------------------------------
// B=8, F=256, T=4096, H=8, D=64, HIDDEN=512, 3*HIDDEN=1536
#define BATCH 8
#define FCH   256
#define TLEN  4096
#define NH    8
#define DH    64
#define HID   512
#define THREEH 1536

typedef __attribute__((ext_vector_type(2))) float v2f;
typedef __attribute__((ext_vector_type(8))) float v8f;

// D = A(16x4) * B(4x16) + C(16x16), all f32, wave32
#define WMMA_F32(accv, av, bv) \
  __builtin_amdgcn_wmma_f32_16x16x4_f32(false, (av), false, (bv), (short)0, (accv), false, false)

// ---------------------------------------------------------------------------
// Kernel 1: QKV projection  Y[b,r,t] = sum_c w_qkv[c,r] * x[b,c,t]
//   Each wave computes a 16(r) x 64(t) strip: one A fragment reused across
//   4 B fragments -> 4 WMMAs per K-step, 2.5 loads/WMMA.
//   r = cd*24 + h*3 + n; elu(x)+1 applied to q,k; scatter to [b,h,cd,t].
// ---------------------------------------------------------------------------
__global__ __launch_bounds__(256) void qkv_proj_kernel(
    const float* __restrict__ x, const float* __restrict__ w_qkv,
    float* __restrict__ qw, float* __restrict__ kw, float* __restrict__ vw) {
  const int lane  = threadIdx.x & 31;
  const int wave  = threadIdx.x >> 5;
  const int lrow  = lane & 15;
  const int lhalf = lane >> 4;

  // strips per batch: 96 (r tiles) * 64 (t groups of 64)
  const int STRIPS_PER_B = (THREEH / 16) * (TLEN / 64);
  int sid = blockIdx.x * 8 + wave;
  int b   = sid / STRIPS_PER_B;
  int rem = sid - b * STRIPS_PER_B;
  int mt  = rem >> 6;          // 0..95
  int ntg = rem & 63;          // 0..63
  int m0  = mt * 16;           // r base
  int n0  = ntg * 64;          // t base (64-wide strip)

  const float* xb = x + (size_t)b * FCH * TLEN;
  const int arow = m0 + lrow;

  v8f acc0 = {}, acc1 = {}, acc2 = {}, acc3 = {};
  for (int kk = 0; kk < FCH; kk += 4) {
    int c0 = kk + 2 * lhalf;
    v2f a;
    a.x = w_qkv[(size_t)c0 * THREEH + arow];
    a.y = w_qkv[(size_t)(c0 + 1) * THREEH + arow];
    const float* xr0 = xb + (size_t)c0 * TLEN + n0 + lrow;
    const float* xr1 = xr0 + TLEN;
    v2f b0, b1, b2, b3;
    b0.x = xr0[0];   b0.y = xr1[0];
    b1.x = xr0[16];  b1.y = xr1[16];
    b2.x = xr0[32];  b2.y = xr1[32];
    b3.x = xr0[48];  b3.y = xr1[48];
    acc0 = WMMA_F32(acc0, a, b0);
    acc1 = WMMA_F32(acc1, a, b1);
    acc2 = WMMA_F32(acc2, a, b2);
    acc3 = WMMA_F32(acc3, a, b3);
  }

#pragma unroll
  for (int i = 0; i < 8; ++i) {
    int r  = m0 + i + 8 * lhalf;
    int nI = r % 3;            // 0=q 1=k 2=v
    int h  = (r / 3) & 7;
    int cd = r / 24;
    float* dst = (nI == 0) ? qw : (nI == 1) ? kw : vw;
    float* row = dst + (((size_t)b * NH + h) * DH + cd) * TLEN + n0 + lrow;
    float f0 = acc0[i], f1 = acc1[i], f2 = acc2[i], f3 = acc3[i];
    if (nI < 2) {              // elu(f)+1
      f0 = (f0 > 0.0f) ? (f0 + 1.0f) : __expf(f0);
      f1 = (f1 > 0.0f) ? (f1 + 1.0f) : __expf(f1);
      f2 = (f2 > 0.0f) ? (f2 + 1.0f) : __expf(f2);
      f3 = (f3 > 0.0f) ? (f3 + 1.0f) : __expf(f3);
    }
    row[0]  = f0;
    row[16] = f1;
    row[32] = f2;
    row[48] = f3;
  }
}

// ---------------------------------------------------------------------------
// Kernel 2: per (b,h):  cntxt[d,e] = sum_t k[d,t]*v[e,t]
//                       norm[d]    = sum_t q[d,t]*k[d,t]
// 512 threads = 16 waves = 16 output tiles of the 64x64 context matrix.
// K-loop loads are adjacent float pairs -> v2f (global_load_b64).
// ---------------------------------------------------------------------------
__global__ __launch_bounds__(512) void context_kernel(
    const float* __restrict__ q, const float* __restrict__ k,
    const float* __restrict__ v, float* __restrict__ cntxt,
    float* __restrict__ norm) {
  int bh = blockIdx.x; // 0..63
  const float* qb = q + (size_t)bh * DH * TLEN;
  const float* kb = k + (size_t)bh * DH * TLEN;
  const float* vb = v + (size_t)bh * DH * TLEN;

  const int lane  = threadIdx.x & 31;
  const int wave  = threadIdx.x >> 5;   // 0..15
  const int lrow  = lane & 15;
  const int lhalf = lane >> 4;
  int m0 = (wave >> 2) * 16;            // d tile
  int n0 = (wave & 3) * 16;             // e tile

  const v2f* krow2 = (const v2f*)(kb + (size_t)(m0 + lrow) * TLEN);
  const v2f* vrow2 = (const v2f*)(vb + (size_t)(n0 + lrow) * TLEN);

  v8f acc = {};
  for (int tt = 0; tt < TLEN; tt += 4) {
    int p = (tt >> 1) + lhalf;          // float2 index: t0 = tt + 2*lhalf
    v2f a  = krow2[p];
    v2f bb = vrow2[p];
    acc = WMMA_F32(acc, a, bb);
  }
#pragma unroll
  for (int i = 0; i < 8; ++i) {
    int d = m0 + i + 8 * lhalf;
    int e = n0 + lrow;
    cntxt[(size_t)bh * (DH * DH) + d * DH + e] = acc[i];
  }

  // norm[d] = sum_t q[d,t]*k[d,t]  (block-cooperative reduction)
  __shared__ float part[512];
  int d   = threadIdx.x & 63;
  int seg = threadIdx.x >> 6;  // 0..7, each covers 512 t
  const v2f* qr2 = (const v2f*)(qb + (size_t)d * TLEN);
  const v2f* kr2 = (const v2f*)(kb + (size_t)d * TLEN);
  float s = 0.f;
  for (int t2 = seg * 256; t2 < seg * 256 + 256; ++t2) {
    v2f qv = qr2[t2], kv = kr2[t2];
    s += qv.x * kv.x + qv.y * kv.y;
  }
  part[threadIdx.x] = s;
  __syncthreads();
  if (threadIdx.x < 64) {
    float tot = 0.f;
#pragma unroll
    for (int j = 0; j < 8; ++j) tot += part[threadIdx.x + j * 64];
    norm[(size_t)bh * DH + threadIdx.x] = tot;
  }
}

// ---------------------------------------------------------------------------
// Kernel 3 (fused): per (b, 16-wide t slab):
//   stage 1: wave w == head h:  P[t,e] = sum_d q[d,t]*cntxt[d,e]
//            val[t, h*64+e] = P * 0.125 / norm[e]   -> LDS (padded rows)
//   stage 2: out[b,o,t] = sum_c val[t,c] * w_out[c,o]   (K=512 from LDS)
// ---------------------------------------------------------------------------
#define LDS_STRIDE 516  // 516 % 64 == 4 -> conflict-free column access
__global__ __launch_bounds__(256) void output_kernel(
    const float* __restrict__ q, const float* __restrict__ cntxt,
    const float* __restrict__ norm, const float* __restrict__ w_out,
    float* __restrict__ out) {
  __shared__ float val_lds[16 * LDS_STRIDE]; // ~33 KB

  int blk   = blockIdx.x;       // B * (T/16) = 2048
  int b     = blk >> 8;
  int tbase = (blk & 255) * 16;

  const int lane  = threadIdx.x & 31;
  const int wave  = threadIdx.x >> 5;  // 0..7 == head
  const int lrow  = lane & 15;
  const int lhalf = lane >> 4;

  // ---- stage 1: P = Q_t^T (16xD) x cntxt (DxE), 4 e-tiles per wave ----
  {
    int h  = wave;
    int bh = b * NH + h;
    const float* qb = q + (size_t)bh * DH * TLEN;
    const float* cx = cntxt + (size_t)bh * (DH * DH);
    const float* nb = norm + (size_t)bh * DH;
    int tglob = tbase + lrow;

    for (int nt = 0; nt < 4; ++nt) {
      int n0 = nt * 16;
      v8f acc = {};
      for (int dd = 0; dd < DH; dd += 4) {
        int d0 = dd + 2 * lhalf;
        v2f a, bb;
        a.x  = qb[(size_t)d0 * TLEN + tglob];
        a.y  = qb[(size_t)(d0 + 1) * TLEN + tglob];
        bb.x = cx[d0 * DH + n0 + lrow];
        bb.y = cx[(d0 + 1) * DH + n0 + lrow];
        acc = WMMA_F32(acc, a, bb);
      }
      int e = n0 + lrow;
      float rn = 0.125f / nb[e];   // /sqrt(64) then /norm[b,h,e]
#pragma unroll
      for (int i = 0; i < 8; ++i) {
        int tl = i + 8 * lhalf;
        val_lds[tl * LDS_STRIDE + h * DH + e] = acc[i] * rn;
      }
    }
  }
  __syncthreads();

  // ---- stage 2: out tile (16 t x 256 o): 16 n-tiles, 2 per wave ----
  const v2f* vl2 = (const v2f*)val_lds;   // LDS A-fragment pairs (8B aligned)
  for (int tt = 0; tt < 2; ++tt) {
    int n0 = (wave * 2 + tt) * 16;
    v8f acc = {};
    for (int cc = 0; cc < HID; cc += 4) {
      int c0 = cc + 2 * lhalf;
      v2f a  = vl2[lrow * (LDS_STRIDE / 2) + (cc >> 1) + lhalf]; // ds_load_b64
      v2f bb;
      bb.x = w_out[(size_t)c0 * FCH + n0 + lrow];
      bb.y = w_out[(size_t)(c0 + 1) * FCH + n0 + lrow];
      acc = WMMA_F32(acc, a, bb);
    }
#pragma unroll
    for (int i = 0; i < 8; ++i) {
      int tg = tbase + i + 8 * lhalf;
      int o  = n0 + lrow;
      out[((size_t)b * FCH + o) * TLEN + tg] = acc[i];
    }
  }
}

// ---------------------------------------------------------------------------
extern "C" void kernel_launch(void* const* d_in, const int* in_sizes, int n_in,
                              void* d_out, int out_size, void* d_ws, size_t ws_size,
                              hipStream_t stream) {
  (void)in_sizes; (void)n_in; (void)out_size; (void)ws_size;
  const float* x     = (const float*)d_in[0];
  const float* w_qkv = (const float*)d_in[1];
  const float* w_out = (const float*)d_in[2];
  float* out = (float*)d_out;
  float* ws  = (float*)d_ws;

  const size_t QS = (size_t)BATCH * NH * DH * TLEN; // 16,777,216 floats
  float* qw = ws;
  float* kw = ws + QS;
  float* vw = ws + 2 * QS;
  float* cx = ws + 3 * QS;                            // B*H*64*64
  float* nm = cx + (size_t)BATCH * NH * DH * DH;      // B*H*64

  // K1: 8 b * 96 r-tiles * 64 t-strips = 49152 strips / 8 waves per block
  qkv_proj_kernel<<<6144, 256, 0, stream>>>(x, w_qkv, qw, kw, vw);
  // K2: one block per (b,h)
  context_kernel<<<BATCH * NH, 512, 0, stream>>>(qw, kw, vw, cx, nm);
  // K3: one block per (b, 16-t slab)
  output_kernel<<<BATCH * (TLEN / 16), 256, 0, stream>>>(qw, cx, nm, w_out, out);
}